// Head_5549097746603
// MI455X (gfx1250) — compile-verified
//
#include <hip/hip_runtime.h>
#include <stdint.h>

#define NB      16
#define NANCH   25200        // 80*80*3 + 40*40*3 + 20*20*3
#define NCLS    80
#define PRE     1000
#define PREP    1024         // padded candidate count (power of 2, one per thread)
#define DET     100
#define SCORE_T 0.25f
#define NMS_T   0.45f
#define MAXSZ   4096.0f
#define IMGSZ   640.0f
#define NBUCKET 4096
#define SUPW    (PREP / 32)  // 32 words of suppression bits per row

typedef float v2f __attribute__((ext_vector_type(2)));
typedef float v8f __attribute__((ext_vector_type(8)));
typedef int   i32x4 __attribute__((vector_size(16)));   // matches builtin param type

// ---- feature detection for CDNA5-specific paths (guarded: compile-safe) ----
#if __has_builtin(__builtin_amdgcn_wmma_f32_16x16x4_f32)
#define USE_WMMA 1
#endif
#if __has_builtin(__builtin_amdgcn_global_load_async_to_lds_b128) && \
    __has_builtin(__builtin_amdgcn_global_load_async_to_lds_b32)  && \
    __has_builtin(__builtin_amdgcn_s_wait_asynccnt)
#define USE_ASYNC_LDS 1
#endif
#define AS1 __attribute__((address_space(1)))
#define AS3 __attribute__((address_space(3)))

#if __has_builtin(__builtin_amdgcn_rcpf)
__device__ __forceinline__ float fastrcp(float x) { return __builtin_amdgcn_rcpf(x); }
#else
__device__ __forceinline__ float fastrcp(float x) { return 1.0f / x; }
#endif

// sigmoid via hardware v_exp + v_rcp (no IEEE divide sequence)
__device__ __forceinline__ float sigm(float x) { return fastrcp(1.0f + __expf(-x)); }

__device__ __forceinline__ void anch_decomp(int a, int& lvl, int& h, int& w, int& aa,
                                            int& W, float& stride) {
    if (a < 19200)      { lvl = 0; W = 80; stride = 8.0f; }
    else if (a < 24000) { lvl = 1; W = 40; stride = 16.0f; a -= 19200; }
    else                { lvl = 2; W = 20; stride = 32.0f; a -= 24000; }
    aa = a % 3; int cell = a / 3; w = cell % W; h = cell / W;
}

// ---------------------------------------------------------------- zero scratch
__global__ void k_zero(unsigned* __restrict__ p, size_t n) {
    size_t i = (size_t)blockIdx.x * blockDim.x + threadIdx.x;
    size_t st = (size_t)gridDim.x * blockDim.x;
    for (; i < n; i += st) p[i] = 0u;
}

// ------------------------------------------------------------------ box decode
__global__ void k_decode(const float* __restrict__ p0, const float* __restrict__ p1,
                         const float* __restrict__ p2, const float* __restrict__ anc,
                         float* __restrict__ boxes) {
    int gid = blockIdx.x * blockDim.x + threadIdx.x;
    if (gid >= NB * NANCH) return;
    int b = gid / NANCH, a = gid % NANCH;
    int lvl, h, w, aa, W; float stride;
    anch_decomp(a, lvl, h, w, aa, W, stride);
    const float* p = (lvl == 0) ? p0 : (lvl == 1 ? p1 : p2);
    size_t off = ((((size_t)b * W + h) * W + w) * 3 + aa) * 85;
    float tx = sigm(p[off + 0]);
    float ty = sigm(p[off + 1]);
    float tw = sigm(p[off + 2]);
    float th = sigm(p[off + 3]);
    float ax = anc[(lvl * 3 + aa) * 2 + 0];
    float ay = anc[(lvl * 3 + aa) * 2 + 1];
    float cx = (2.0f * tx - 0.5f + (float)w) * stride;
    float cy = (2.0f * ty - 0.5f + (float)h) * stride;
    float bw = 4.0f * tw * tw * ax;
    float bh = 4.0f * th * th * ay;
    float x1 = fminf(fmaxf(cx - 0.5f * bw, 0.0f), IMGSZ);
    float y1 = fminf(fmaxf(cy - 0.5f * bh, 0.0f), IMGSZ);
    float x2 = fminf(fmaxf(cx + 0.5f * bw, 0.0f), IMGSZ);
    float y2 = fminf(fmaxf(cy + 0.5f * bh, 0.0f), IMGSZ);
    float* o = boxes + (size_t)gid * 4;
    o[0] = x1; o[1] = y1; o[2] = x2; o[3] = y2;
}

// -------------------------------------------- radix-select pass 1: histogram
__global__ void k_hist(const float* __restrict__ p0, const float* __restrict__ p1,
                       const float* __restrict__ p2, unsigned* __restrict__ hist) {
    int gid = blockIdx.x * blockDim.x + threadIdx.x;
    if (gid >= NB * NANCH) return;
    int b = gid / NANCH, a = gid % NANCH;
    int lvl, h, w, aa, W; float stride;
    anch_decomp(a, lvl, h, w, aa, W, stride);
    const float* p = (lvl == 0) ? p0 : (lvl == 1 ? p1 : p2);
    size_t off = ((((size_t)b * W + h) * W + w) * 3 + aa) * 85;
    float obj = sigm(p[off + 4]);
    if (obj <= SCORE_T) return;
    for (int c = 0; c < NCLS; ++c) {
        float s = obj * sigm(p[off + 5 + c]);
        if (s > SCORE_T) {
            int bk = (int)(s * (float)NBUCKET);
            bk = bk > (NBUCKET - 1) ? (NBUCKET - 1) : bk;
            atomicAdd(&hist[(size_t)b * NBUCKET + bk], 1u);
        }
    }
}

// ------------------------------------------ radix-select pass 2: find cutoff
__global__ void k_thresh(const unsigned* __restrict__ hist, unsigned* __restrict__ thr) {
    int b = threadIdx.x;
    if (b >= NB) return;
    unsigned c = 0; unsigned t = 0;
    for (int k = NBUCKET - 1; k >= 0; --k) {
        c += hist[(size_t)b * NBUCKET + k];
        if (c >= PRE) { t = (unsigned)k; break; }
    }
    thr[b] = t;
}

// ------------------------------------------ radix-select pass 3: compaction
__global__ void k_compact(const float* __restrict__ p0, const float* __restrict__ p1,
                          const float* __restrict__ p2, const float* __restrict__ boxes,
                          const unsigned* __restrict__ thr, unsigned* __restrict__ cnt,
                          float* __restrict__ cScore, unsigned* __restrict__ cLab,
                          float* __restrict__ cBox) {
    int gid = blockIdx.x * blockDim.x + threadIdx.x;
    if (gid >= NB * NANCH) return;
    int b = gid / NANCH, a = gid % NANCH;
    int lvl, h, w, aa, W; float stride;
    anch_decomp(a, lvl, h, w, aa, W, stride);
    const float* p = (lvl == 0) ? p0 : (lvl == 1 ? p1 : p2);
    size_t off = ((((size_t)b * W + h) * W + w) * 3 + aa) * 85;
    float obj = sigm(p[off + 4]);
    if (obj <= SCORE_T) return;
    unsigned th = thr[b];
    const float* bx = boxes + (size_t)gid * 4;
    for (int c = 0; c < NCLS; ++c) {
        float s = obj * sigm(p[off + 5 + c]);
        if (s > SCORE_T) {
            int bk = (int)(s * (float)NBUCKET);
            bk = bk > (NBUCKET - 1) ? (NBUCKET - 1) : bk;
            if ((unsigned)bk >= th) {
                unsigned pos = atomicAdd(&cnt[b], 1u);
                if (pos < PRE) {
                    size_t ci = (size_t)b * PREP + pos;
                    cScore[ci] = s;
                    cLab[ci] = (unsigned)c;
                    float* d = cBox + ci * 4;
                    d[0] = bx[0]; d[1] = bx[1]; d[2] = bx[2]; d[3] = bx[3];
                }
            }
        }
    }
}

// --------------------------------------------------- matrix-NMS, one block/img
__device__ __forceinline__ float areaBox(const float* sb, int i) {
    return (sb[i * 4 + 2] - sb[i * 4 + 0]) * (sb[i * 4 + 3] - sb[i * 4 + 1]);
}

__global__ __launch_bounds__(PREP)
void k_nms(const float* __restrict__ cScore, const float* __restrict__ cBox,
           const unsigned* __restrict__ cLab, unsigned* __restrict__ sup,
           const float* __restrict__ scaleF, float* __restrict__ out) {
    __shared__ float sb[PREP * 4];      // class-offset boxes
    __shared__ float ss[PREP];          // scores
    __shared__ unsigned char alive[PREP];
    __shared__ float s_rv[32];
    __shared__ int   s_ri[32];
    __shared__ float s_bv;
    __shared__ int   s_bj;
    __shared__ int   keepI[DET];
    __shared__ float keepS[DET];

    const int img  = blockIdx.x;
    const int t    = threadIdx.x;
    const int wave = t >> 5, lane = t & 31;
    unsigned* supImg = sup + (size_t)img * PREP * SUPW;

    // ---- stage candidates into LDS (async DMA path on gfx1250) ----
    {
        const float* gb = cBox + ((size_t)img * PREP + t) * 4;
        const float* gs = cScore + (size_t)img * PREP + t;
#ifdef USE_ASYNC_LDS
        __builtin_amdgcn_global_load_async_to_lds_b128(
            (AS1 i32x4*)const_cast<float*>(gb), (AS3 i32x4*)&sb[t * 4], 0, 0);
        __builtin_amdgcn_global_load_async_to_lds_b32(
            (AS1 int*)const_cast<float*>(gs), (AS3 int*)&ss[t], 0, 0);
        __builtin_amdgcn_s_wait_asynccnt(0);
#else
        sb[t * 4 + 0] = gb[0]; sb[t * 4 + 1] = gb[1];
        sb[t * 4 + 2] = gb[2]; sb[t * 4 + 3] = gb[3];
        ss[t] = gs[0];
#endif
        alive[t] = 1;
    }
    __syncthreads();
    {   // class-aware offset: obx = box + label*4096 (pad slots: label 0, box 0)
        float loff = (float)cLab[(size_t)img * PREP + t] * MAXSZ;
        sb[t * 4 + 0] += loff; sb[t * 4 + 1] += loff;
        sb[t * 4 + 2] += loff; sb[t * 4 + 3] += loff;
    }
    __syncthreads();

    // ---- build 1024x1024 suppression bitmask; one 16x16 tile per wave/iter ----
    // area_i + area_j for the whole tile via one V_WMMA_F32_16X16X4_F32:
    //   A(16x4) = [area_i, 1, 0, 0],  B(4x16) = [1; area_j; 0; 0]
    const int lm = lane & 15, halfSel = lane >> 4;
    for (int tt = wave; tt < (PREP / 16) * (PREP / 16); tt += 32) {
        const int ti = tt >> 6, tj = tt & 63;
        const int iRow = ti * 16 + lm, jCol = tj * 16 + lm;
        const float ai = areaBox(sb, iRow);
        const float aj = areaBox(sb, jCol);
        v8f acc = {0.f, 0.f, 0.f, 0.f, 0.f, 0.f, 0.f, 0.f};
#ifdef USE_WMMA
        v2f av, bv;
        av[0] = halfSel ? 0.0f : ai;    // A: K=0 col = area_i (K=2 col zero)
        av[1] = halfSel ? 0.0f : 1.0f;  //    K=1 col = ones   (K=3 col zero)
        bv[0] = halfSel ? 0.0f : 1.0f;  // B: K=0 row = ones
        bv[1] = halfSel ? 0.0f : aj;    //    K=1 row = area_j
        acc = __builtin_amdgcn_wmma_f32_16x16x4_f32(
            false, av, false, bv, (short)0, acc, false, false);
#else
        for (int r = 0; r < 8; ++r)
            acc[r] = areaBox(sb, ti * 16 + r + (halfSel ? 8 : 0)) + aj;
#endif
        const float jx1 = sb[jCol * 4 + 0], jy1 = sb[jCol * 4 + 1];
        const float jx2 = sb[jCol * 4 + 2], jy2 = sb[jCol * 4 + 3];
        for (int r = 0; r < 8; ++r) {
            const int i = ti * 16 + r + (halfSel ? 8 : 0);
            const float iw = fmaxf(fminf(sb[i*4+2], jx2) - fmaxf(sb[i*4+0], jx1), 0.0f);
            const float ih = fmaxf(fminf(sb[i*4+3], jy2) - fmaxf(sb[i*4+1], jy1), 0.0f);
            const float inter = iw * ih;
            // iou > T  <=>  inter > T*(union)   (union = areaSum - inter + eps > 0)
            // one FMA + one compare instead of a full IEEE divide sequence
            const float rhs = NMS_T * (acc[r] - inter + 1e-7f);
            if (inter > rhs)
                atomicOr(&supImg[(size_t)i * SUPW + (jCol >> 5)], 1u << (jCol & 31));
        }
    }
    __threadfence();
    __syncthreads();

    // ---- greedy selection: 100x (block argmax + bitmask kill) ----
    for (int det = 0; det < DET; ++det) {
        float v = alive[t] ? ss[t] : 0.0f;
        int id = t;
        for (int offm = 16; offm; offm >>= 1) {
            float ov = __shfl_xor(v, offm, 32);
            int   oi = __shfl_xor(id, offm, 32);
            if (ov > v) { v = ov; id = oi; }
        }
        if (lane == 0) { s_rv[wave] = v; s_ri[wave] = id; }
        __syncthreads();
        if (wave == 0) {
            float v2 = s_rv[lane]; int i2 = s_ri[lane];
            for (int offm = 16; offm; offm >>= 1) {
                float ov = __shfl_xor(v2, offm, 32);
                int   oi = __shfl_xor(i2, offm, 32);
                if (ov > v2) { v2 = ov; i2 = oi; }
            }
            if (lane == 0) { s_bv = v2; s_bj = i2; keepI[det] = i2; keepS[det] = v2; }
        }
        __syncthreads();
        const float bvv = s_bv; const int bj = s_bj;
        if (bvv > 0.0f) {
            unsigned wbits = supImg[(size_t)bj * SUPW + (t >> 5)];
            if ((wbits >> (t & 31)) & 1u) alive[t] = 0;
        }
        __syncthreads();
    }

    // ---- emit 100x6 rows ----
    if (t < DET) {
        const int j = keepI[t]; const float v = keepS[t];
        const float rs = fastrcp(scaleF[img]);   // scale is a multiply, not 4 divides
        float* o = out + ((size_t)img * DET + t) * 6;
        if (v > 0.0f) {
            const float* bx = cBox + ((size_t)img * PREP + j) * 4;
            o[0] = bx[0] * rs; o[1] = bx[1] * rs;
            o[2] = bx[2] * rs; o[3] = bx[3] * rs;
            o[4] = v;
            o[5] = (float)cLab[(size_t)img * PREP + j];
        } else {
            o[0] = 0.f; o[1] = 0.f; o[2] = 0.f; o[3] = 0.f;
            o[4] = 0.f; o[5] = -1.0f;
        }
    }
}

// ----------------------------------------------------------------------------
extern "C" void kernel_launch(void* const* d_in, const int* in_sizes, int n_in,
                              void* d_out, int out_size, void* d_ws, size_t ws_size,
                              hipStream_t stream) {
    const float* p0  = (const float*)d_in[0];
    const float* p1  = (const float*)d_in[1];
    const float* p2  = (const float*)d_in[2];
    const float* anc = (const float*)d_in[3];
    const float* scl = (const float*)d_in[4];
    float* out = (float*)d_out;

    char* ws = (char*)d_ws;
    size_t off = 0;
    auto alloc = [&](size_t bytes) {
        size_t o = off; off += (bytes + 255) & ~(size_t)255; return o;
    };
    float*    boxes  = (float*)(ws + alloc((size_t)NB * NANCH * 4 * sizeof(float)));
    size_t zeroStart = off;
    unsigned* hist   = (unsigned*)(ws + alloc((size_t)NB * NBUCKET * 4));
    unsigned* thr    = (unsigned*)(ws + alloc((size_t)NB * 4));
    unsigned* cnt    = (unsigned*)(ws + alloc((size_t)NB * 4));
    float*    cScore = (float*)(ws + alloc((size_t)NB * PREP * 4));
    unsigned* cLab   = (unsigned*)(ws + alloc((size_t)NB * PREP * 4));
    float*    cBox   = (float*)(ws + alloc((size_t)NB * PREP * 16));
    unsigned* sup    = (unsigned*)(ws + alloc((size_t)NB * PREP * SUPW * 4));
    size_t zeroWords = (off - zeroStart) / 4;

    k_zero<<<1024, 256, 0, stream>>>((unsigned*)(ws + zeroStart), zeroWords);

    const int total = NB * NANCH;
    const int blk = 256;
    const int grd = (total + blk - 1) / blk;
    k_decode <<<grd, blk, 0, stream>>>(p0, p1, p2, anc, boxes);
    k_hist   <<<grd, blk, 0, stream>>>(p0, p1, p2, hist);
    k_thresh <<<1, 32, 0, stream>>>(hist, thr);
    k_compact<<<grd, blk, 0, stream>>>(p0, p1, p2, boxes, thr, cnt, cScore, cLab, cBox);
    k_nms    <<<NB, PREP, 0, stream>>>(cScore, cBox, cLab, sup, scl, out);
}